// DefaultCRSegmentor_2508260901469
// MI455X (gfx1250) — compile-verified
//
#include <hip/hip_runtime.h>

typedef __attribute__((ext_vector_type(2))) float v2f;
typedef __attribute__((ext_vector_type(8))) float v8f;

#define NUM_CLS       20
#define NUM_CLUSTERS  65536
#define EPSN          1e-4f

struct F3 { float x, y, z; };

__device__ __forceinline__ float swap16(float v) {
  // ds_swizzle group-of-32: xor=0x10, or=0x00, and=0x1f  -> offset 0x401F
  return __int_as_float(
      __builtin_amdgcn_ds_swizzle(__float_as_int(v), 0x401F));
}

// ---------------- workspace layout (floats/ints) -------------------------
// cls_sum : 3*C*K floats   (later overwritten in-place with per-(c,l) means)
// cls_cnt :   C*K floats
// geo_ctr : 3*C   floats
// pure    :   C   ints
// lmin    :   C   ints
// lmax    :   C   ints
// accum   :   8   floats (accum[0]=sum smoothL1, accum[1]=sum (1-cos))

__global__ void crs_init_ws(float* __restrict__ cls_data,
                            int* __restrict__ lmin, int* __restrict__ lmax,
                            float* __restrict__ accum) {
  const long long stride = (long long)gridDim.x * blockDim.x;
  const long long tid    = (long long)blockIdx.x * blockDim.x + threadIdx.x;
  const long long total  = 4LL * NUM_CLUSTERS * NUM_CLS;  // cls_sum + cls_cnt
  for (long long i = tid; i < total; i += stride) cls_data[i] = 0.0f;
  for (long long i = tid; i < NUM_CLUSTERS; i += stride) {
    lmin[i] = 0x7FFFFFFF;
    lmax[i] = (int)0x80000000;
  }
  if (tid < 8) accum[tid] = 0.0f;
}

__global__ void crs_scatter(const F3* __restrict__ grid,
                            const int* __restrict__ cluster,
                            const int* __restrict__ label,
                            float* __restrict__ cls_sum,
                            float* __restrict__ cls_cnt,
                            int* __restrict__ lmin, int* __restrict__ lmax,
                            long long n) {
  const long long stride = (long long)gridDim.x * blockDim.x;
  for (long long i = (long long)blockIdx.x * blockDim.x + threadIdx.x;
       i < n; i += stride) {
    const int c = cluster[i];
    const int l = label[i];
    const int f = c * NUM_CLS + l;
    const F3 g = grid[i];                     // global_load_b96
    atomicAdd(&cls_sum[3 * f + 0], g.x);
    atomicAdd(&cls_sum[3 * f + 1], g.y);
    atomicAdd(&cls_sum[3 * f + 2], g.z);
    atomicAdd(&cls_cnt[f], 1.0f);
    atomicMin(&lmin[c], l);
    atomicMax(&lmax[c], l);
  }
}

// per cluster: geo center = sum of its K class slots; convert class sums to
// means in place; purity flag from label min/max.
__global__ void crs_finalize(float* __restrict__ cls_sum,
                             const float* __restrict__ cls_cnt,
                             float* __restrict__ geo_ctr,
                             int* __restrict__ pure,
                             const int* __restrict__ lmin,
                             const int* __restrict__ lmax) {
  const int stride = gridDim.x * blockDim.x;
  for (int c = blockIdx.x * blockDim.x + threadIdx.x; c < NUM_CLUSTERS;
       c += stride) {
    float sx = 0.f, sy = 0.f, sz = 0.f, cnt = 0.f;
    for (int l = 0; l < NUM_CLS; ++l) {
      const int f = c * NUM_CLS + l;
      const float ct = cls_cnt[f];
      const float x = cls_sum[3 * f + 0];
      const float y = cls_sum[3 * f + 1];
      const float z = cls_sum[3 * f + 2];
      sx += x; sy += y; sz += z; cnt += ct;
      const float inv = 1.0f / fmaxf(ct, 1.0f);
      cls_sum[3 * f + 0] = x * inv;
      cls_sum[3 * f + 1] = y * inv;
      cls_sum[3 * f + 2] = z * inv;
    }
    const float inv = 1.0f / fmaxf(cnt, 1.0f);
    geo_ctr[3 * c + 0] = sx * inv;
    geo_ctr[3 * c + 1] = sy * inv;
    geo_ctr[3 * c + 2] = sz * inv;
    pure[c] = (lmin[c] == lmax[c]) ? 1 : 0;
  }
}

__device__ __forceinline__ void crs_point_loss(
    const F3* __restrict__ pred, const F3* __restrict__ grid,
    const int* __restrict__ cluster, const int* __restrict__ label,
    const float* __restrict__ cls_mean, const float* __restrict__ geo_ctr,
    const int* __restrict__ pure, long long i, bool valid,
    float& l1, float& dd) {
  const int c = cluster[i];
  const int l = label[i];
  const int f = c * NUM_CLS + l;
  const F3 g = grid[i];
  const bool p = pure[c] != 0;
  const float tx = p ? geo_ctr[3 * c + 0] : cls_mean[3 * f + 0];
  const float ty = p ? geo_ctr[3 * c + 1] : cls_mean[3 * f + 1];
  const float tz = p ? geo_ctr[3 * c + 2] : cls_mean[3 * f + 2];
  const float tox = tx - g.x, toy = ty - g.y, toz = tz - g.z;
  const F3 pr = pred[i];
  const float dx = pr.x - tox, dy = pr.y - toy, dz = pr.z - toz;
  const float ax = fabsf(dx), ay = fabsf(dy), az = fabsf(dz);
  l1 = (ax < 1.0f ? 0.5f * dx * dx : ax - 0.5f) +
       (ay < 1.0f ? 0.5f * dy * dy : ay - 0.5f) +
       (az < 1.0f ? 0.5f * dz * dz : az - 0.5f);
  const float np = fmaxf(sqrtf(pr.x * pr.x + pr.y * pr.y + pr.z * pr.z), EPSN);
  const float nt = fmaxf(sqrtf(tox * tox + toy * toy + toz * toz), EPSN);
  float cosv = (pr.x * tox + pr.y * toy + pr.z * toz) / (np * nt);
  cosv = fminf(fmaxf(cosv, -1.0f), 1.0f);
  dd = 1.0f - cosv;
  l1 = valid ? l1 : 0.0f;
  dd = valid ? dd : 0.0f;
}

// Loss pass: per-lane pairs of point losses feed the A matrix of
// V_WMMA_F32_16X16X4_F32 with B = ones, C = running accumulator, so the
// matrix core performs the running reduction (D = A*1 + D, f32 throughout).
__global__ void crs_loss(const F3* __restrict__ pred,
                         const F3* __restrict__ grid,
                         const int* __restrict__ cluster,
                         const int* __restrict__ label,
                         const float* __restrict__ cls_mean,
                         const float* __restrict__ geo_ctr,
                         const int* __restrict__ pure,
                         float* __restrict__ accum, long long n) {
  const long long T    = (long long)gridDim.x * blockDim.x;
  const long long gtid = (long long)blockIdx.x * blockDim.x + threadIdx.x;
  const long long iters = (n + 2 * T - 1) / (2 * T);  // uniform trip count

  v8f accL = {};
  v8f accD = {};
  v2f ones;
  ones[0] = 1.0f;
  ones[1] = 1.0f;

  for (long long it = 0; it < iters; ++it) {
    const long long i0 = gtid + it * 2 * T;
    const long long i1 = i0 + T;
    const bool v0 = i0 < n;
    const bool v1 = i1 < n;
    float l1a, dda, l1b, ddb;
    crs_point_loss(pred, grid, cluster, label, cls_mean, geo_ctr, pure,
                   v0 ? i0 : 0, v0, l1a, dda);
    crs_point_loss(pred, grid, cluster, label, cls_mean, geo_ctr, pure,
                   v1 ? i1 : 0, v1, l1b, ddb);
    v2f aL; aL[0] = l1a; aL[1] = l1b;
    v2f aD; aD[0] = dda; aD[1] = ddb;
    // D = A x ones(4x16) + C : row-sums accumulate, EXEC is all-ones here.
    accL = __builtin_amdgcn_wmma_f32_16x16x4_f32(false, aL, false, ones,
                                                 (short)0, accL, false, false);
    accD = __builtin_amdgcn_wmma_f32_16x16x4_f32(false, aD, false, ones,
                                                 (short)0, accD, false, false);
  }

  // C/D layout: VGPR j, lanes 0-15 = row M=j, lanes 16-31 = row M=8+j, and
  // every column is identical (B was all-ones). Per-lane sum of the 8 VGPRs
  // gives sum(z[0..7]) on lanes 0-15 and sum(z[8..15]) on lanes 16-31; one
  // half-swap + add yields the full wave total on every lane.
  float sL = 0.f, sD = 0.f;
#pragma unroll
  for (int j = 0; j < 8; ++j) { sL += accL[j]; sD += accD[j]; }
  sL += swap16(sL);
  sD += swap16(sD);

  __shared__ float redL[8];
  __shared__ float redD[8];
  const int wave = threadIdx.x >> 5;
  const int lane = threadIdx.x & 31;
  if (lane == 0) { redL[wave] = sL; redD[wave] = sD; }
  __syncthreads();
  if (threadIdx.x == 0) {
    float tL = 0.f, tD = 0.f;
    const int nw = blockDim.x >> 5;
    for (int w = 0; w < nw; ++w) { tL += redL[w]; tD += redD[w]; }
    atomicAdd(&accum[0], tL);
    atomicAdd(&accum[1], tD);
  }
}

__global__ void crs_writeout(const float* __restrict__ accum,
                             float* __restrict__ out, long long n) {
  if (threadIdx.x == 0 && blockIdx.x == 0) {
    out[0] = accum[0] / (float)(3LL * n);  // mean over N*3 elements
    out[1] = accum[1] / (float)n;          // mean over N points
  }
}

extern "C" void kernel_launch(void* const* d_in, const int* in_sizes, int n_in,
                              void* d_out, int out_size, void* d_ws,
                              size_t ws_size, hipStream_t stream) {
  const F3*  pred_off = (const F3*)d_in[0];
  const F3*  grid     = (const F3*)d_in[1];
  const int* cluster  = (const int*)d_in[2];
  const int* label    = (const int*)d_in[3];
  const long long n = (long long)in_sizes[2];  // cluster has N elements

  const long long CK = (long long)NUM_CLUSTERS * NUM_CLS;
  float* cls_sum = (float*)d_ws;                  // 3*C*K
  float* cls_cnt = cls_sum + 3 * CK;              // C*K
  float* geo_ctr = cls_cnt + CK;                  // 3*C
  int*   pure    = (int*)(geo_ctr + 3LL * NUM_CLUSTERS);  // C
  int*   lmin    = pure + NUM_CLUSTERS;           // C
  int*   lmax    = lmin + NUM_CLUSTERS;           // C
  float* accum   = (float*)(lmax + NUM_CLUSTERS); // 8

  crs_init_ws<<<2048, 256, 0, stream>>>(cls_sum, lmin, lmax, accum);
  crs_scatter<<<4096, 256, 0, stream>>>(grid, cluster, label, cls_sum, cls_cnt,
                                        lmin, lmax, n);
  crs_finalize<<<256, 256, 0, stream>>>(cls_sum, cls_cnt, geo_ctr, pure, lmin,
                                        lmax);
  crs_loss<<<2048, 256, 0, stream>>>(pred_off, grid, cluster, label, cls_sum,
                                     geo_ctr, pure, accum, n);
  crs_writeout<<<1, 32, 0, stream>>>(accum, (float*)d_out, n);
}